// SparseTrellisLinear_678604833216
// MI455X (gfx1250) — compile-verified
//
#include <hip/hip_runtime.h>

// ---------------- CDNA5 vector operand types ----------------
typedef __attribute__((ext_vector_type(8)))  float  v8f;
typedef __attribute__((ext_vector_type(16))) __bf16 v16bf;   // SWMMAC A: 16x32 stored bf16 (expands to 16x64)
typedef __attribute__((ext_vector_type(32))) __bf16 v32bf;   // SWMMAC B: dense 64x16 bf16

// ---------------- problem dims (match reference) ----------------
#define K_DENSE   4096
#define N_OUT     11008
#define M_TOK     4096
#define KS        2048      // sparse K (2:4 keeps half)
#define TK        128       // sparse-k tiles of 16
#define TNT       688       // n tiles of 16
#define PB        96        // packed bytes per 16x16 tile (3-bit codes)
#define NGROUPS   1024      // K/4 sparsity groups
#define KC        64        // k-chunks: 64 dense K (= 32 stored) per SWMMAC
#define TMT       256       // m tiles of 16

// workspace layout (bytes)
#define A_BYTES   ((size_t)TNT * KC * 32 * 16 * 2)   // 45,088,768  bf16 A values, SWMMAC lane layout
#define I_BYTES   ((size_t)TNT * KC * 32 * 4)        //  5,636,096  u32 index words, SWMMAC lane layout
#define X_BYTES   ((size_t)TMT * KC * 32 * 32 * 2)   // 33,554,432  bf16 B values, SWMMAC lane layout

// CDNA5 swmmac builtin (signature verified by round-1 codegen:
// lowers to v_swmmac_f32_16x16x64_bf16 vD[8], vA[8], vB[16], vIdx).
#define SWMMAC_BF16(A, B, C, IDX) \
  __builtin_amdgcn_swmmac_f32_16x16x64_bf16(false, (A), false, (B), (C), (int)(IDX), false, false)

// =====================================================================
// Stage 1a: dequantize trellis codes -> bf16 A values in SWMMAC layout
//   entry = ((nt*KC + kc)*32 + lane), 16 bf16 per entry (stored-K order
//   per the ISA 16-bit A-matrix 16x32 table: lanes 0-15 hold stored K
//   0..7,16..23; lanes 16-31 hold 8..15,24..31).
// =====================================================================
__global__ __launch_bounds__(256) void stl_pack_a(
    const int* __restrict__ packed, const int* __restrict__ meta,
    const float* __restrict__ grid, const float* __restrict__ scales,
    const float* __restrict__ su, const float* __restrict__ sv,
    __bf16* __restrict__ Aval)
{
  int t = blockIdx.x * 256 + threadIdx.x;            // one thread per bf16 value
  const int total = TNT * KC * 32 * 16;              // 22,544,384
  if (t >= total) return;
  int j    =  t        & 15;                         // position within lane's 16 values
  int lane = (t >> 4)  & 31;
  int kc   = (t >> 9)  & 63;
  int nt   =  t >> 15;

  // stored-K index within chunk per ISA A-layout
  int s;
  if (lane < 16) s = (j < 8) ? j        : (j + 8);   // 0..7, 16..23
  else           s = (j < 8) ? (j + 8)  : (j + 16);  // 8..15, 24..31
  int ks = kc * 32 + s;                              // global stored sparse row
  int n  = nt * 16 + (lane & 15);                    // A row M = output column n

  // --- decode 3-bit code (little-endian bit stream, tile-major) ---
  int kt = ks >> 4, lk = ks & 15, ln = n & 15;
  int c  = lk * 16 + ln;
  int bitpos = 3 * c;
  long pbase = ((long)kt * TNT + nt) * PB;
  unsigned b0 = (unsigned)packed[pbase + (bitpos >> 3)] & 0xFFu;
  int by1 = (bitpos >> 3) + 1;
  unsigned b1 = (by1 < PB) ? ((unsigned)packed[pbase + by1] & 0xFFu) : 0u;
  unsigned code = ((b0 | (b1 << 8)) >> (bitpos & 7)) & 7u;

  // --- 2:4 metadata: which dense k this stored value occupies ---
  int g = ks >> 1;
  unsigned w   = (unsigned)meta[(long)(g >> 3) * N_OUT + n];
  unsigned nib = (w >> (4 * (g & 7))) & 0xFu;
  int idx = (ks & 1) ? (int)((nib >> 2) & 3u) : (int)(nib & 3u);
  int dk  = 4 * g + idx;                             // dense K position

  float v = grid[code] * scales[(long)(dk >> 7) * N_OUT + n] * su[dk] * sv[n];
  Aval[t] = (__bf16)v;
}

// =====================================================================
// Stage 1b: repack 2:4 nibbles -> SWMMAC index VGPR words
//   lane L holds 8 nibbles for row M=L%16; lanes 0-15 = groups 0..7 of
//   the chunk, lanes 16-31 = groups 8..15 (ISA 7.12.4 pseudocode).
//   The reference nibble (idx1<<2)|idx0 matches the ISA encoding.
// =====================================================================
__global__ __launch_bounds__(256) void stl_pack_idx(
    const int* __restrict__ meta, unsigned* __restrict__ Aidx)
{
  int t = blockIdx.x * 256 + threadIdx.x;
  const int total = TNT * KC * 32;                   // 1,409,024
  if (t >= total) return;
  int lane =  t        & 31;
  int kc   = (t >> 5)  & 63;
  int nt   =  t >> 11;
  int n    = nt * 16 + (lane & 15);
  int cg0  = (lane < 16) ? 0 : 8;
  unsigned out = 0;
#pragma unroll
  for (int i = 0; i < 8; ++i) {
    int g = kc * 16 + cg0 + i;                       // global group; g&7 == i
    unsigned w   = (unsigned)meta[(long)(g >> 3) * N_OUT + n];
    unsigned nib = (w >> (4 * i)) & 0xFu;
    out |= nib << (4 * i);
  }
  Aidx[t] = out;
}

// =====================================================================
// Stage 1c: x (f32, row-major [M,K]) -> bf16 B operand, dense 64x16 layout
//   B[k][m] = x[m][k]; lane holds column m = lane%16.
//   lanes 0-15: dense k 0..15 then 32..47 ; lanes 16-31: 16..31 then 48..63.
// =====================================================================
__global__ __launch_bounds__(256) void stl_pack_x(
    const float* __restrict__ x, __bf16* __restrict__ Bx)
{
  int t = blockIdx.x * 256 + threadIdx.x;
  const int total = TMT * KC * 32 * 32;              // 16,777,216
  if (t >= total) return;
  int j    =  t        & 31;
  int lane = (t >> 5)  & 31;
  int kc   = (t >> 10) & 63;
  int mt   =  t >> 16;
  int kk;
  if (lane < 16) kk = (j < 16) ? j        : (j + 16);
  else           kk = (j < 16) ? (j + 16) : (j + 32);
  int m = mt * 16 + (lane & 15);
  Bx[t] = (__bf16)x[(long)m * K_DENSE + kc * 64 + kk];
}

// =====================================================================
// Stage 2: sparse GEMM.  out^T tile = A(16n x 64k sparse) x B(64k x 16m).
//   Block = 8 waves; wave (wn,wm) owns 4 n-tiles x 2 m-tiles (8 v8f accs).
//   Grid = (N/128, M/128) = (86, 32).  64 k-chunks, 8 swmmac per chunk.
//   All operand loads are per-lane-contiguous b128s; intra-block reuse
//   (4x on A, 2x on B) is absorbed by the WGP$ / 192MB L2.
// =====================================================================
__global__ __launch_bounds__(256) void stl_swmmac_gemm(
    const __bf16* __restrict__ Aval, const unsigned* __restrict__ Aidx,
    const __bf16* __restrict__ Bx, float* __restrict__ out)
{
  const int lane = threadIdx.x & 31;
  const int wv   = threadIdx.x >> 5;
  const int wn   = wv & 1;           // 2 waves across n
  const int wm   = wv >> 1;          // 4 waves across m
  const int nt0  = blockIdx.x * 8 + wn * 4;
  const int mt0  = blockIdx.y * 8 + wm * 2;

  const v16bf* __restrict__ Ap = (const v16bf*)Aval;   // 32B per entry
  const v32bf* __restrict__ Bp = (const v32bf*)Bx;     // 64B per entry

  v8f acc[4][2];
#pragma unroll
  for (int i = 0; i < 4; ++i)
#pragma unroll
    for (int jm = 0; jm < 2; ++jm)
      acc[i][jm] = (v8f){0.f, 0.f, 0.f, 0.f, 0.f, 0.f, 0.f, 0.f};

  for (int kc = 0; kc < KC; ++kc) {
    v16bf a[4];
    unsigned ia[4];
#pragma unroll
    for (int i = 0; i < 4; ++i) {
      long e = ((long)(nt0 + i) * KC + kc) * 32 + lane;
      a[i]  = Ap[e];
      ia[i] = Aidx[e];
      if (kc + 1 < KC) __builtin_prefetch(&Ap[e + 32], 0, 1);  // next k-chunk (entry stride 32)
    }
    v32bf b[2];
#pragma unroll
    for (int jm = 0; jm < 2; ++jm) {
      long e = ((long)(mt0 + jm) * KC + kc) * 32 + lane;
      b[jm] = Bp[e];
      if (kc + 1 < KC) __builtin_prefetch(&Bp[e + 32], 0, 1);  // next k-chunk
    }
#pragma unroll
    for (int i = 0; i < 4; ++i)
#pragma unroll
      for (int jm = 0; jm < 2; ++jm)
        acc[i][jm] = SWMMAC_BF16(a[i], b[jm], acc[i][jm], ia[i]);
  }

  // D layout: lane l, vgpr v -> row(M=n) = v + 8*(l>=16), col(N=m) = l%16.
  const int row = lane & 15;
  const int hi  = lane >> 4;
#pragma unroll
  for (int i = 0; i < 4; ++i)
#pragma unroll
    for (int jm = 0; jm < 2; ++jm) {
      int  m    = (mt0 + jm) * 16 + row;
      long base = (long)m * N_OUT + (long)(nt0 + i) * 16 + hi * 8;
      float4 f0, f1;
      f0.x = acc[i][jm][0]; f0.y = acc[i][jm][1]; f0.z = acc[i][jm][2]; f0.w = acc[i][jm][3];
      f1.x = acc[i][jm][4]; f1.y = acc[i][jm][5]; f1.z = acc[i][jm][6]; f1.w = acc[i][jm][7];
      *(float4*)(out + base)     = f0;
      *(float4*)(out + base + 4) = f1;
    }
}

// =====================================================================
extern "C" void kernel_launch(void* const* d_in, const int* in_sizes, int n_in,
                              void* d_out, int out_size, void* d_ws, size_t ws_size,
                              hipStream_t stream) {
  const float* x      = (const float*)d_in[0];   // [M,K]
  const float* scales = (const float*)d_in[1];   // [K/128, N]
  const float* su     = (const float*)d_in[2];   // [K]
  const float* sv     = (const float*)d_in[3];   // [N]
  const float* grid   = (const float*)d_in[4];   // [8]
  const int*   packed = (const int*)d_in[5];     // [TK,TN,96] byte values
  const int*   meta   = (const int*)d_in[6];     // [K/32, N] packed nibbles
  float*       out    = (float*)d_out;

  char*     ws   = (char*)d_ws;
  __bf16*   Aval = (__bf16*)ws;
  unsigned* Aidx = (unsigned*)(ws + A_BYTES);
  __bf16*   Bx   = (__bf16*)(ws + A_BYTES + I_BYTES);

  {
    const int total = TNT * KC * 32 * 16;
    stl_pack_a<<<(total + 255) / 256, 256, 0, stream>>>(packed, meta, grid, scales, su, sv, Aval);
  }
  {
    const int total = TNT * KC * 32;
    stl_pack_idx<<<(total + 255) / 256, 256, 0, stream>>>(meta, Aidx);
  }
  {
    const int total = TMT * KC * 32 * 32;
    stl_pack_x<<<(total + 255) / 256, 256, 0, stream>>>(x, Bx);
  }
  {
    dim3 g(TNT / 8, TMT / 8);   // (86, 32)
    stl_swmmac_gemm<<<g, 256, 0, stream>>>(Aval, Aidx, Bx, out);
  }
}